// HashTriEmbedder_85830626443280
// MI455X (gfx1250) — compile-verified
//
#include <hip/hip_runtime.h>
#include <stdint.h>

// ---------------------------------------------------------------------------
// Tri-plane multires hash encoding (instant-NGP style), MI455X / gfx1250.
// Memory-gather-bound: ~201MB table (~fits 192MB L2), 192 random 8B gathers
// per point. No matrix structure -> no WMMA; CDNA5 value comes from the
// async LDS->global store path (ASYNCcnt) + deep unrolled gather batching.
// ---------------------------------------------------------------------------

#define NLVL   16
#define NPLANE 3
#define FDIM   2
#define TSIZE  (1u << 19)
#define HPRIME 2654435761u
#define BLOCK  128
#define OUT_DIM 96            // NLVL * NPLANE * FDIM
#define ROWP   100            // padded LDS row stride (dwords): 36*L mod 64
                              // spreads b128 blocks over all bank groups
                              // (2-way min conflict vs 16-way at stride 96)

// floor(16 * 2^(l/3)) for l = 0..15 (GROWTH = 32^(1/15) = 2^(1/3))
constexpr float RESF[NLVL] = {
    16.f, 20.f, 25.f, 32.f, 40.f, 50.f, 64.f, 80.f,
    101.f, 128.f, 161.f, 203.f, 256.f, 322.f, 406.f, 512.f};

// -------- async LDS -> global store (CDNA5 ASYNCcnt path) ------------------
typedef int v4i __attribute__((ext_vector_type(4)));
typedef __attribute__((address_space(1))) v4i gv4i;   // global int4
typedef __attribute__((address_space(3))) v4i lv4i;   // LDS int4

#if defined(__has_builtin)
#if __has_builtin(__builtin_amdgcn_global_store_async_from_lds_b128)
#define HAVE_ASYNC_B128 1
#endif
#endif

__device__ __forceinline__ void async_store_b128(float* g, const float* l) {
#ifdef HAVE_ASYNC_B128
    // Verified signature: (as(1) int4*, as(3) int4*, imm offset, imm cpol)
    __builtin_amdgcn_global_store_async_from_lds_b128(
        (gv4i*)g, (lv4i*)l, /*offset=*/0, /*cpol=*/0);
#else
    uint64_t ga = (uint64_t)g;
    uint32_t la = (uint32_t)(uintptr_t)(__attribute__((address_space(3))) const float*)l;
    asm volatile("global_store_async_from_lds_b128 %0, %1, off"
                 :: "v"(ga), "v"(la) : "memory");
#endif
}

// ---------------------------------------------------------------------------

extern "C" __global__ __launch_bounds__(BLOCK, 1)
void HashTriEmbedder_85830626443280_kernel(const float* __restrict__ x,
                                           const float* __restrict__ emb,
                                           float* __restrict__ out)
{
    __shared__ __align__(16) float tile[BLOCK * ROWP];      // 50 KB

    const int tid = threadIdx.x;
    const int n   = blockIdx.x * BLOCK + tid;

    // Coalesced b96 load of the point
    const float3 pt = *(const float3*)(x + 3ull * (unsigned)n);
    const float  P[3] = {pt.x, pt.y, pt.z};

    const float2* __restrict__ e2 = (const float2*)emb;   // [3*16*T] float2
    float* row = &tile[tid * ROWP];

#pragma unroll
    for (int l = 0; l < NLVL; ++l) {
        const float resf  = RESF[l];
        const float grid  = 1.0f / resf;   // constant-folded (fp32, matches ref)
        const float rgrid = 1.0f / grid;   // constant-folded

#pragma unroll
        for (int p = 0; p < NPLANE; ++p) {
            const int a0 = (p == 2) ? 1 : 0;
            const int a1 = (p == 0) ? 1 : 2;
            const float x0 = P[a0], x1 = P[a1];

            const float t0 = x0 * rgrid;
            const float t1 = x1 * rgrid;
            const float f0 = floorf(t0);
            const float f1 = floorf(t1);
            const float w0 = (x0 - f0 * grid) * rgrid;   // (xp - vmin)/grid
            const float w1 = (x1 - f1 * grid) * rgrid;

            const uint32_t c0 = (uint32_t)(int)f0;
            const uint32_t c1 = (uint32_t)(int)f1;
            const uint32_t base = (uint32_t)(p * NLVL + l) * TSIZE;

            // corners: (0,0) (0,1) (1,0) (1,1);  h = cx ^ (cy * PRIME)
            const uint32_t m0 =  c1        * HPRIME;
            const uint32_t m1 = (c1 + 1u)  * HPRIME;
            const uint32_t h00 = ( c0       ^ m0) & (TSIZE - 1u);
            const uint32_t h01 = ( c0       ^ m1) & (TSIZE - 1u);
            const uint32_t h10 = ((c0 + 1u) ^ m0) & (TSIZE - 1u);
            const uint32_t h11 = ((c0 + 1u) ^ m1) & (TSIZE - 1u);

            // 4 random b64 gathers (L2-resident table)
            const float2 v00 = e2[base + h00];
            const float2 v01 = e2[base + h01];
            const float2 v10 = e2[base + h10];
            const float2 v11 = e2[base + h11];

            const float om0 = 1.0f - w0;
            const float om1 = 1.0f - w1;
            const float c0x = v00.x * om0 + v10.x * w0;
            const float c0y = v00.y * om0 + v10.y * w0;
            const float c1x = v01.x * om0 + v11.x * w0;
            const float c1y = v01.y * om0 + v11.y * w0;

            const int c = l * 6 + p * 2;                 // [N, L, 3, F] layout
            row[c + 0] = c0x * om1 + c1x * w1;
            row[c + 1] = c0y * om1 + c1y * w1;
        }
    }

    __syncthreads();

    // Cooperative, fully coalesced write-out via async LDS->global b128.
    // Global side stays perfectly linear; LDS side maps flat float4 index
    // q -> (point r = q/24, col c = (q%24)*4) in the padded tile.
    float* outBase = out + (size_t)blockIdx.x * (BLOCK * OUT_DIM);
#pragma unroll
    for (int it = 0; it < OUT_DIM / 4; ++it) {           // 24 iterations
        const int q = it * BLOCK + tid;                  // float4 index
        const int r = q / (OUT_DIM / 4);                 // /24: const division
        const int c = (q - r * (OUT_DIM / 4)) * 4;       // dword col, 16B aligned
        async_store_b128(outBase + 4 * q, &tile[r * ROWP + c]);
    }
    // s_endpgm performs an implicit wait-idle (ASYNCcnt drains before exit).
}

extern "C" void kernel_launch(void* const* d_in, const int* in_sizes, int n_in,
                              void* d_out, int out_size, void* d_ws, size_t ws_size,
                              hipStream_t stream) {
    const float* x   = (const float*)d_in[0];   // [N,3] f32
    const float* emb = (const float*)d_in[1];   // [3,16,2^19,2] f32
    float* out = (float*)d_out;                 // [N,96] f32

    const int npts   = in_sizes[0] / 3;         // 1048576
    const int blocks = npts / BLOCK;            // 8192 (N divisible by 128)

    hipLaunchKernelGGL(HashTriEmbedder_85830626443280_kernel,
                       dim3(blocks), dim3(BLOCK), 0, stream, x, emb, out);
}